// SelectiveSSMExpert_16793322127758
// MI455X (gfx1250) — compile-verified
//
#include <hip/hip_runtime.h>

// ---------------------------------------------------------------------------
// Selective SSM for gfx1250 (MI455X). All GEMMs via V_WMMA_F32_16X16X4_F32
// (exact f32 math; workload is HBM-bound at ~450MB / 23.3 TB/s, so f32 WMMA
// already saturates). Scan parallelized via chunked prefix recurrence.
// ---------------------------------------------------------------------------

typedef __attribute__((ext_vector_type(2))) float v2f;
typedef __attribute__((ext_vector_type(8))) float v8f;

#define DIMX   1024
#define NSTATE 256
#define SELH   512
#define NB     4
#define NT     4096
#define NROWS  (NB * NT)      // 16384 flattened (b,t) rows
#define CH     128            // scan chunk length
#define NCH    (NT / CH)      // 32 chunks per sequence

__device__ __forceinline__ float fast_sigmoid(float x) {
    return 1.0f / (1.0f + __expf(-x));
}
__device__ __forceinline__ float fast_tanh(float x) {
    x = fminf(15.0f, fmaxf(-15.0f, x));
    float e = __expf(2.0f * x);
    return (e - 1.0f) / (e + 1.0f);
}

__device__ __forceinline__ v8f wmma_f32(v2f a, v2f b, v8f c) {
    // D = A(16x4) x B(4x16) + C(16x16), f32, wave32
    return __builtin_amdgcn_wmma_f32_16x16x4_f32(
        /*neg_a=*/false, a, /*neg_b=*/false, b,
        /*c_mod=*/(short)0, c, /*reuse_a=*/false, /*reuse_b=*/false);
}

// Shared main loop: one wave computes a 16(M) x 64(N) tile.
// A-layout per ISA: lane L holds A[M = L%16][K = v + 2*(L/16)] in VGPR v.
// B-layout:         lane L holds B[K = v + 2*(L/16)][N = L%16] in VGPR v.
// For C = X * W^T this means A and B load with the SAME pattern: row index
// from lane%16, two contiguous K elements at offset k + 2*(lane/16).
template <int K>
__device__ __forceinline__ void gemm_tile_16x64(const float* __restrict__ ap,
                                                const float* __restrict__ bp,
                                                v8f acc[4]) {
    const float* bp0 = bp;
    const float* bp1 = bp + (size_t)16 * K;
    const float* bp2 = bp + (size_t)32 * K;
    const float* bp3 = bp + (size_t)48 * K;
#pragma unroll 8
    for (int k = 0; k < K; k += 4) {
        v2f a  = *(const v2f*)(ap  + k);
        v2f b0 = *(const v2f*)(bp0 + k);
        v2f b1 = *(const v2f*)(bp1 + k);
        v2f b2 = *(const v2f*)(bp2 + k);
        v2f b3 = *(const v2f*)(bp3 + k);
        acc[0] = wmma_f32(a, b0, acc[0]);
        acc[1] = wmma_f32(a, b1, acc[1]);
        acc[2] = wmma_f32(a, b2, acc[2]);
        acc[3] = wmma_f32(a, b3, acc[3]);
    }
}

// ---------------------------------------------------------------------------
// Kernel 1: u = x @ W_in^T  and  sh = silu(x @ W_sel_in^T), fused (x read once
// per N-tile; weights + x are L2 resident). grid = (NROWS/64, 768/64), blk=128.
// ---------------------------------------------------------------------------
__global__ __launch_bounds__(128) void ssm_proj_kernel(
    const float* __restrict__ X, const float* __restrict__ Win,
    const float* __restrict__ Wsel, float* __restrict__ U,
    float* __restrict__ SH) {
    const int wave = threadIdx.x >> 5;
    const int lane = threadIdx.x & 31;
    const int half = lane >> 4;
    const int lm   = lane & 15;
    const int mbase = (blockIdx.x * 4 + wave) * 16;
    const int nbase = blockIdx.y * 64;  // [0, 768): first 256 -> u, rest -> sh

    const float* Wp;
    int nloc;
    if (nbase < NSTATE) { Wp = Win;  nloc = nbase; }
    else                { Wp = Wsel; nloc = nbase - NSTATE; }

    const float* ap = X  + (size_t)(mbase + lm) * DIMX + 2 * half;
    const float* bp = Wp + (size_t)(nloc  + lm) * DIMX + 2 * half;

    v8f acc[4] = {v8f{}, v8f{}, v8f{}, v8f{}};
    gemm_tile_16x64<DIMX>(ap, bp, acc);

#pragma unroll
    for (int sub = 0; sub < 4; ++sub) {
        const int col = nbase + sub * 16 + lm;
#pragma unroll
        for (int v = 0; v < 8; ++v) {
            const int row = mbase + v + 8 * half;
            float val = acc[sub][v];
            if (col < NSTATE) {
                U[(size_t)row * NSTATE + col] = val;
            } else {
                SH[(size_t)row * SELH + (col - NSTATE)] = val * fast_sigmoid(val);
            }
        }
    }
}

// ---------------------------------------------------------------------------
// Kernel 2: gates = sh @ W_sel_out^T with fused activations; writes the four
// scan operands directly (raw gates never hit memory).
//   q0: A   = sigmoid(g)          q1: BU  = tanh(g) * u
//   q2: C   = tanh(g)             q3: DGU = d_param * sigmoid(g) * u
// grid = (NROWS/64, 1024/64), blk = 128.
// ---------------------------------------------------------------------------
__global__ __launch_bounds__(128) void ssm_gates_kernel(
    const float* __restrict__ SH, const float* __restrict__ Wso,
    const float* __restrict__ U, const float* __restrict__ dpar,
    float* __restrict__ Aa, float* __restrict__ BU, float* __restrict__ Cg,
    float* __restrict__ DGU) {
    const int wave = threadIdx.x >> 5;
    const int lane = threadIdx.x & 31;
    const int half = lane >> 4;
    const int lm   = lane & 15;
    const int mbase = (blockIdx.x * 4 + wave) * 16;
    const int nbase = blockIdx.y * 64;  // [0, 1024)

    const float* ap = SH  + (size_t)(mbase + lm) * SELH + 2 * half;
    const float* bp = Wso + (size_t)(nbase + lm) * SELH + 2 * half;

    v8f acc[4] = {v8f{}, v8f{}, v8f{}, v8f{}};
    gemm_tile_16x64<SELH>(ap, bp, acc);

#pragma unroll
    for (int sub = 0; sub < 4; ++sub) {
        const int g  = nbase + sub * 16 + lm;
        const int q  = g >> 8;
        const int nn = g & 255;
#pragma unroll
        for (int v = 0; v < 8; ++v) {
            const int row = mbase + v + 8 * half;
            const size_t ridx = (size_t)row * NSTATE + nn;
            float val = acc[sub][v];
            if (q == 0)      Aa[ridx]  = fast_sigmoid(val);
            else if (q == 1) BU[ridx]  = fast_tanh(val) * U[ridx];
            else if (q == 2) Cg[ridx]  = fast_tanh(val);
            else             DGU[ridx] = dpar[nn] * fast_sigmoid(val) * U[ridx];
        }
    }
}

// ---------------------------------------------------------------------------
// Chunked parallel scan: h_t = a_t * h_{t-1} + bu_t.
// pass1: per-(b,chunk,n) compute (P = prod a, S = chunk-local scan end).
// pass2: per-(b,n) combine 32 chunk carries -> H0 (incoming h per chunk).
// pass3: replay chunks with correct H0, fuse y = C*h + DGU.
// ---------------------------------------------------------------------------
__global__ __launch_bounds__(256) void ssm_scan_pass1(
    const float* __restrict__ Aa, const float* __restrict__ BU,
    float* __restrict__ P, float* __restrict__ S) {
    const int b = blockIdx.x / NCH;
    const int chunk = blockIdx.x % NCH;
    const int n = threadIdx.x;
    size_t base = (size_t)(b * NT + chunk * CH) * NSTATE + n;
    float h = 0.f, p = 1.f;
#pragma unroll 4
    for (int i = 0; i < CH; ++i) {
        float a  = Aa[base + (size_t)i * NSTATE];
        float bu = BU[base + (size_t)i * NSTATE];
        h = fmaf(a, h, bu);
        p *= a;
    }
    const size_t cidx = (size_t)(b * NCH + chunk) * NSTATE + n;
    P[cidx] = p;
    S[cidx] = h;
}

__global__ __launch_bounds__(256) void ssm_scan_pass2(
    const float* __restrict__ P, const float* __restrict__ S,
    float* __restrict__ H0) {
    const int b = blockIdx.x;
    const int n = threadIdx.x;
    float h = 0.f;
#pragma unroll
    for (int c = 0; c < NCH; ++c) {
        const size_t idx = (size_t)(b * NCH + c) * NSTATE + n;
        H0[idx] = h;
        h = fmaf(P[idx], h, S[idx]);
    }
}

__global__ __launch_bounds__(256) void ssm_scan_pass3(
    const float* __restrict__ Aa, const float* __restrict__ BU,
    const float* __restrict__ Cg, const float* __restrict__ DGU,
    const float* __restrict__ H0, float* __restrict__ Y) {
    const int b = blockIdx.x / NCH;
    const int chunk = blockIdx.x % NCH;
    const int n = threadIdx.x;
    float h = H0[(size_t)(b * NCH + chunk) * NSTATE + n];
    size_t base = (size_t)(b * NT + chunk * CH) * NSTATE + n;
#pragma unroll 4
    for (int i = 0; i < CH; ++i) {
        const size_t idx = base + (size_t)i * NSTATE;
        h = fmaf(Aa[idx], h, BU[idx]);
        Y[idx] = fmaf(Cg[idx], h, DGU[idx]);
    }
}

// ---------------------------------------------------------------------------
// Kernel 6: out = y @ W_out^T.  grid = (NROWS/64, 1024/64), blk = 128.
// ---------------------------------------------------------------------------
__global__ __launch_bounds__(128) void ssm_out_kernel(
    const float* __restrict__ Y, const float* __restrict__ Wout,
    float* __restrict__ Out) {
    const int wave = threadIdx.x >> 5;
    const int lane = threadIdx.x & 31;
    const int half = lane >> 4;
    const int lm   = lane & 15;
    const int mbase = (blockIdx.x * 4 + wave) * 16;
    const int nbase = blockIdx.y * 64;

    const float* ap = Y    + (size_t)(mbase + lm) * NSTATE + 2 * half;
    const float* bp = Wout + (size_t)(nbase + lm) * NSTATE + 2 * half;

    v8f acc[4] = {v8f{}, v8f{}, v8f{}, v8f{}};
    gemm_tile_16x64<NSTATE>(ap, bp, acc);

#pragma unroll
    for (int sub = 0; sub < 4; ++sub) {
        const int col = nbase + sub * 16 + lm;
#pragma unroll
        for (int v = 0; v < 8; ++v) {
            const int row = mbase + v + 8 * half;
            Out[(size_t)row * DIMX + col] = acc[sub][v];
        }
    }
}

// ---------------------------------------------------------------------------
extern "C" void kernel_launch(void* const* d_in, const int* in_sizes, int n_in,
                              void* d_out, int out_size, void* d_ws,
                              size_t ws_size, hipStream_t stream) {
    (void)in_sizes; (void)n_in; (void)out_size; (void)ws_size;

    const float* x     = (const float*)d_in[0];  // [4,4096,1024]
    const float* Win   = (const float*)d_in[1];  // [256,1024]
    const float* Wsel  = (const float*)d_in[2];  // [512,1024]
    const float* Wso   = (const float*)d_in[3];  // [1024,512]
    const float* Wout  = (const float*)d_in[4];  // [1024,256]
    const float* dpar  = (const float*)d_in[5];  // [256]
    float* out = (float*)d_out;                  // [4,4096,1024]

    // Workspace layout (~113 MB). Y aliases U (U dead after gates kernel).
    char* ws = (char*)d_ws;
    const size_t SZ_N = (size_t)NROWS * NSTATE * sizeof(float);  // 16 MB
    const size_t SZ_H = (size_t)NROWS * SELH * sizeof(float);    // 32 MB
    const size_t SZ_C = (size_t)NB * NCH * NSTATE * sizeof(float);  // 128 KB
    size_t off = 0;
    float* U   = (float*)(ws + off); off += SZ_N;
    float* SH  = (float*)(ws + off); off += SZ_H;
    float* Aa  = (float*)(ws + off); off += SZ_N;
    float* BU  = (float*)(ws + off); off += SZ_N;
    float* Cg  = (float*)(ws + off); off += SZ_N;
    float* DGU = (float*)(ws + off); off += SZ_N;
    float* P   = (float*)(ws + off); off += SZ_C;
    float* S   = (float*)(ws + off); off += SZ_C;
    float* H0  = (float*)(ws + off); off += SZ_C;
    float* Y   = U;  // safe alias: U only consumed inside ssm_gates_kernel

    // 1) fused input projections
    ssm_proj_kernel<<<dim3(NROWS / 64, (NSTATE + SELH) / 64), 128, 0, stream>>>(
        x, Win, Wsel, U, SH);
    // 2) gate GEMM + activations -> scan operands
    ssm_gates_kernel<<<dim3(NROWS / 64, (4 * NSTATE) / 64), 128, 0, stream>>>(
        SH, Wso, U, dpar, Aa, BU, Cg, DGU);
    // 3-5) chunked parallel scan + fused y
    ssm_scan_pass1<<<dim3(NB * NCH), 256, 0, stream>>>(Aa, BU, P, S);
    ssm_scan_pass2<<<dim3(NB), 256, 0, stream>>>(P, S, H0);
    ssm_scan_pass3<<<dim3(NB * NCH), 256, 0, stream>>>(Aa, BU, Cg, DGU, H0, Y);
    // 6) output projection
    ssm_out_kernel<<<dim3(NROWS / 64, DIMX / 64), 128, 0, stream>>>(Y, Wout, out);
}